// DurationConstrainedHMM_18279380812096
// MI455X (gfx1250) — compile-verified
//
#include <hip/hip_runtime.h>
#include <hip/hip_bf16.h>

typedef __attribute__((ext_vector_type(2))) float v2f;
typedef __attribute__((ext_vector_type(8))) float v8f;

#define BB 64
#define TT 2000
#define SS 128
#define FF 80

// ---------------------------------------------------------------------------
// Phase 1: emission net.  One wave per 16-row tile of the (B*T, F) input.
// GEMM1 (K=80) and GEMM2 (K=128) via V_WMMA_F32_16X16X4_F32, H staged in LDS.
// ---------------------------------------------------------------------------
__global__ __launch_bounds__(256) void emis_kernel(
    const float* __restrict__ obs, const float* __restrict__ W1,
    const float* __restrict__ b1,  const float* __restrict__ W2,
    const float* __restrict__ b2,  float* __restrict__ emis)
{
  extern __shared__ float lds[];            // 8 waves * 16 * 132 floats
  const int lane = threadIdx.x & 31;
  const int wave = threadIdx.x >> 5;
  const int row0 = (blockIdx.x * 8 + wave) * 16;

  const int m  = lane & 15;                 // A-matrix row (M)
  const int kk = (lane >> 4) << 1;          // K sub-offset: 0 (lanes 0-15) / 2 (16-31)
  const int c  = lane & 15;                 // B/C/D column within 16-tile
  const int hi = (lane >> 4) * 8;           // C/D row offset for upper lanes

  float* Hs = lds + wave * (16 * 132);      // padded stride: conflict-free re-read

  // ---- GEMM1: H = relu(obs @ W1 + b1), 16 x 128 ----
  v8f acc[8];
#pragma unroll
  for (int n = 0; n < 8; ++n)
#pragma unroll
    for (int i = 0; i < 8; ++i) acc[n][i] = 0.f;

  for (int k = 0; k < 20; ++k) {            // K = 80 = 20 * 4
    const int k0 = k * 4;
    v2f a = *(const v2f*)(obs + (size_t)(row0 + m) * FF + k0 + kk);
#pragma unroll
    for (int n = 0; n < 8; ++n) {
      v2f bf;
      bf.x = W1[(k0 + kk) * SS + n * 16 + c];
      bf.y = W1[(k0 + kk + 1) * SS + n * 16 + c];
      acc[n] = __builtin_amdgcn_wmma_f32_16x16x4_f32(
          false, a, false, bf, (short)0, acc[n], false, false);
    }
  }
  // bias + relu, spill H tile to LDS (row-major, stride 132 floats)
#pragma unroll
  for (int n = 0; n < 8; ++n) {
    float bb = b1[n * 16 + c];
#pragma unroll
    for (int i = 0; i < 8; ++i) {
      float h = acc[n][i] + bb;
      Hs[(hi + i) * 132 + n * 16 + c] = h > 0.f ? h : 0.f;
    }
  }
  asm volatile("s_wait_dscnt 0" ::: "memory");   // wave-local LDS RAW fence

  // ---- GEMM2: logits = H @ W2 + b2, K = 128 = 32 * 4 ----
  v8f acc2[8];
#pragma unroll
  for (int n = 0; n < 8; ++n)
#pragma unroll
    for (int i = 0; i < 8; ++i) acc2[n][i] = 0.f;

  for (int k = 0; k < 32; ++k) {
    const int k0 = k * 4;
    v2f a = *(const v2f*)(Hs + m * 132 + k0 + kk);
#pragma unroll
    for (int n = 0; n < 8; ++n) {
      v2f bf;
      bf.x = W2[(k0 + kk) * SS + n * 16 + c];
      bf.y = W2[(k0 + kk + 1) * SS + n * 16 + c];
      acc2[n] = __builtin_amdgcn_wmma_f32_16x16x4_f32(
          false, a, false, bf, (short)0, acc2[n], false, false);
    }
  }
#pragma unroll
  for (int n = 0; n < 8; ++n) {
    float bb = b2[n * 16 + c];
#pragma unroll
    for (int i = 0; i < 8; ++i) acc2[n][i] += bb;
  }

  // ---- row-wise log-softmax over 128 cols; rows live in 16-lane halves ----
#pragma unroll
  for (int i = 0; i < 8; ++i) {
    float mx = -3.4e38f;
#pragma unroll
    for (int n = 0; n < 8; ++n) mx = fmaxf(mx, acc2[n][i]);
#pragma unroll
    for (int d = 1; d <= 8; d <<= 1) mx = fmaxf(mx, __shfl_xor(mx, d, 32));
    float se = 0.f;
#pragma unroll
    for (int n = 0; n < 8; ++n) se += __expf(acc2[n][i] - mx);
#pragma unroll
    for (int d = 1; d <= 8; d <<= 1) se += __shfl_xor(se, d, 32);
    float lse = mx + __logf(se);
    const int row = row0 + hi + i;
#pragma unroll
    for (int n = 0; n < 8; ++n)
      emis[(size_t)row * SS + n * 16 + c] = acc2[n][i] - lse;
  }
}

// ---------------------------------------------------------------------------
__global__ void init_bar(unsigned int* bar) { if (threadIdx.x == 0) *bar = 0u; }

// ---------------------------------------------------------------------------
// Phase 2: Viterbi scan. 16 WGs (cluster-sized) x 512 threads; thread = (b,s).
// trans staged via global_load_async_to_lds_b128; per step: duration partials
// -> cluster barrier (+ atomic fallback) -> penalties -> fold pen_switch into
// delta (dps) -> pure max-plus inner loop with -inf diagonal; self-transition
// handled as a register-only candidate with exact first-max tie-break.
// ---------------------------------------------------------------------------
__global__ __launch_bounds__(512) void viterbi_scan(
    const float* __restrict__ emis, const float* __restrict__ trans,
    int* __restrict__ pmax, int* __restrict__ pmin,
    unsigned int* __restrict__ bar, float* __restrict__ finaldelta,
    unsigned char* __restrict__ psi)
{
  extern __shared__ float smem[];
  float*  lt     = smem;                        // [128*128] log-transition (diag = -inf)
  float*  dps    = lt + 128 * 128;              // [4*128] delta + pen_switch
  float2* pen2   = (float2*)(dps + 512);        // [128] (pen_self, pen_switch)
  int*    durtmp = (int*)(pen2 + 128);          // [4*128]

  const int tid = threadIdx.x;
  const int wg  = blockIdx.x;                   // 0..15
  const int bl  = tid >> 7;                     // 0..3 local batch
  const int s   = tid & 127;                    // state
  const int b   = wg * 4 + bl;                  // global batch

  // ---- async-copy raw transition_logits (64KB) into LDS (ASYNCcnt path) ----
  {
#pragma unroll
    for (int it = 0; it < 8; ++it) {
      unsigned ldsoff = (unsigned)(tid * 16 + it * 8192);   // byte offset, lt @ 0
      unsigned long long ga = (unsigned long long)trans + ldsoff;
      asm volatile("global_load_async_to_lds_b128 %0, %1, off"
                   :: "v"(ldsoff), "v"(ga) : "memory");
    }
    asm volatile("s_wait_asynccnt 0x0" ::: "memory");
  }
  __syncthreads();

  // in-place transform: lt = log(softmax(row) + 1e-8); then poison diagonal
  if (tid < 128) {
    float* row = lt + tid * 128;
    float mx = -3.4e38f;
    for (int j = 0; j < 128; ++j) mx = fmaxf(mx, row[j]);
    float se = 0.f;
    for (int j = 0; j < 128; ++j) se += __expf(row[j] - mx);
    float inv = 1.f / se;
    for (int j = 0; j < 128; ++j)
      row[j] = __logf(__expf(row[j] - mx) * inv + 1e-8f);
    row[tid] = -3.0e38f;                         // exclude p==s from generic max
  }

  float delta = emis[(size_t)b * TT * SS + s] - 4.852030263919617f; // ln(128)
  int dur = 1;
  __syncthreads();

  for (int t = 1; t < TT; ++t) {
    const int buf = t & 1;

    // 1) per-WG duration partials -> L2 (double-buffered by step parity)
    durtmp[bl * 128 + s] = dur;
    __syncthreads();
    if (bl == 0) {
      int mx = durtmp[s], mn = mx;
#pragma unroll
      for (int k = 1; k < 4; ++k) {
        int v = durtmp[k * 128 + s];
        mx = v > mx ? v : mx;  mn = v < mn ? v : mn;
      }
      __hip_atomic_store(&pmax[(buf * 16 + wg) * 128 + s], mx,
                         __ATOMIC_RELAXED, __HIP_MEMORY_SCOPE_AGENT);
      __hip_atomic_store(&pmin[(buf * 16 + wg) * 128 + s], mn,
                         __ATOMIC_RELAXED, __HIP_MEMORY_SCOPE_AGENT);
    }
    __builtin_amdgcn_fence(__ATOMIC_RELEASE, "agent");
    __syncthreads();
    // hardware cluster barrier (one wave signals; NOP if not cluster-dispatched)
    if (tid < 32) __builtin_amdgcn_s_cluster_barrier();
    // monotonic atomic barrier: correctness fallback under plain dispatch
    if (tid == 0) {
      __hip_atomic_fetch_add(bar, 1u, __ATOMIC_ACQ_REL, __HIP_MEMORY_SCOPE_AGENT);
      const unsigned int target = 16u * (unsigned)t;
      while (__hip_atomic_load(bar, __ATOMIC_ACQUIRE, __HIP_MEMORY_SCOPE_AGENT) < target)
        __builtin_amdgcn_s_sleep(1);
    }
    __syncthreads();
    __builtin_amdgcn_fence(__ATOMIC_ACQUIRE, "agent");

    // 2) global dur max/min per prev-state -> penalties (tid < 128)
    if (bl == 0) {
      int gmx = -2147483647, gmn = 2147483647;
      for (int w = 0; w < 16; ++w) {
        int vx = __hip_atomic_load(&pmax[(buf * 16 + w) * 128 + s],
                                   __ATOMIC_RELAXED, __HIP_MEMORY_SCOPE_AGENT);
        int vn = __hip_atomic_load(&pmin[(buf * 16 + w) * 128 + s],
                                   __ATOMIC_RELAXED, __HIP_MEMORY_SCOPE_AGENT);
        gmx = vx > gmx ? vx : gmx;  gmn = vn < gmn ? vn : gmn;
      }
      int a1 = gmx + 1 - 30; if (a1 < 0) a1 = 0;   // MAX_D
      int a2 = 3 - gmn;      if (a2 < 0) a2 = 0;   // MIN_D
      pen2[s] = make_float2(-0.1f * (float)a1, -0.1f * (float)a2);
    }
    __syncthreads();

    // 3) fold pen_switch into delta: dps[b][p] = delta[b][p] + pen_switch[p]
    const float2 myp = pen2[s];                  // (.x = pen_self, .y = pen_switch)
    dps[bl * 128 + s] = delta + myp.y;
    __syncthreads();

    // 4) generic max-plus over p != s (diag poisoned), ascending first-max
    const float* dps_b = dps + bl * 128;
    float best = -3.4e38f; int barg = 0;
    for (int p = 0; p < 128; p += 4) {
      const float4 d4 = *(const float4*)(dps_b + p);   // ds_load_b128 broadcast
      float sc0 = d4.x + lt[(p + 0) * 128 + s];
      float sc1 = d4.y + lt[(p + 1) * 128 + s];
      float sc2 = d4.z + lt[(p + 2) * 128 + s];
      float sc3 = d4.w + lt[(p + 3) * 128 + s];
      if (sc0 > best) { best = sc0; barg = p; }
      if (sc1 > best) { best = sc1; barg = p + 1; }
      if (sc2 > best) { best = sc2; barg = p + 2; }
      if (sc3 > best) { best = sc3; barg = p + 3; }
    }
    // self-transition candidate (register-only), exact first-max merge
    const float scs = delta + myp.x;
    if (scs > best || (scs == best && s < barg)) { best = scs; barg = s; }

    float e = emis[((size_t)b * TT + t) * SS + s];
    if (t + 1 < TT)
      __builtin_prefetch(&emis[((size_t)b * TT + t + 1) * SS + s], 0, 0);
    delta = best + e;
    psi[((size_t)(t - 1) * BB + b) * SS + s] = (unsigned char)barg;
    dur = (barg == s) ? dur + 1 : 1;
  }
  finaldelta[b * 128 + s] = delta;
}

// ---------------------------------------------------------------------------
// Phase 3: backtrack, one thread per batch.
// ---------------------------------------------------------------------------
__global__ __launch_bounds__(64) void backtrack(
    const float* __restrict__ finaldelta, const unsigned char* __restrict__ psi,
    int* __restrict__ out)
{
  const int b = threadIdx.x;
  float best = -3.4e38f; int st = 0;
  for (int s = 0; s < 128; ++s) {
    float v = finaldelta[b * 128 + s];
    if (v > best) { best = v; st = s; }
  }
  out[b * TT + (TT - 1)] = st;
  for (int j = TT - 2; j >= 0; --j) {
    st = psi[((size_t)j * BB + b) * SS + st];
    out[b * TT + j] = st;
  }
}

// ---------------------------------------------------------------------------
extern "C" void kernel_launch(void* const* d_in, const int* in_sizes, int n_in,
                              void* d_out, int out_size, void* d_ws, size_t ws_size,
                              hipStream_t stream) {
  const float* obs   = (const float*)d_in[0];
  const float* W1    = (const float*)d_in[1];
  const float* b1    = (const float*)d_in[2];
  const float* W2    = (const float*)d_in[3];
  const float* b2    = (const float*)d_in[4];
  const float* trans = (const float*)d_in[5];
  int* out = (int*)d_out;

  char* ws = (char*)d_ws;
  float* emis = (float*)ws;
  size_t off = (size_t)BB * TT * SS * sizeof(float);          // 65,536,000 B
  int* pmax = (int*)(ws + off);  off += 2 * 16 * 128 * sizeof(int);
  int* pmin = (int*)(ws + off);  off += 2 * 16 * 128 * sizeof(int);
  unsigned int* bar = (unsigned int*)(ws + off); off += 256;
  float* finaldelta = (float*)(ws + off); off += (size_t)BB * SS * sizeof(float);
  unsigned char* psi = (unsigned char*)(ws + off);            // 16,375,808 B

  // Phase 1: 8000 tiles of 16 rows, 8 waves per block
  emis_kernel<<<1000, 256, 8 * 16 * 132 * sizeof(float), stream>>>(
      obs, W1, b1, W2, b2, emis);

  // Phase 2
  init_bar<<<1, 64, 0, stream>>>(bar);
  size_t smem = (128 * 128 + 4 * 128) * sizeof(float)
              + 128 * sizeof(float2) + 4 * 128 * sizeof(int);
  viterbi_scan<<<16, 512, smem, stream>>>(emis, trans, pmax, pmin, bar,
                                          finaldelta, psi);

  // Phase 3
  backtrack<<<1, 64, 0, stream>>>(finaldelta, psi, out);
}